// GNNEncoder_28630251995475
// MI455X (gfx1250) — compile-verified
//
#include <hip/hip_runtime.h>

#define N_NODES 50000
#define N_EDGES 800000
#define D_IN    128
#define D_HID   256
#define D_OUT   256

typedef __attribute__((ext_vector_type(2))) float v2f;
typedef __attribute__((ext_vector_type(8))) float v8f;

// ---------------------------------------------------------------------------
// Hardware fp32 atomic add at device scope (L2 RMW), no return value.
// Forces GLOBAL_ATOMIC_ADD_F32 (no CAS-loop fallback). Tracked by STOREcnt;
// S_ENDPGM performs an implicit wait-idle, so no manual drain is needed.
// ---------------------------------------------------------------------------
__device__ __forceinline__ void atomic_add_f32_dev(float* p, float v) {
    asm volatile("global_atomic_add_f32 %0, %1, off scope:SCOPE_DEV"
                 :
                 : "v"(p), "v"(v)
                 : "memory");
}

// ---------------------------------------------------------------------------
// Zero a float buffer (grid-stride).
// ---------------------------------------------------------------------------
__global__ void zero_f32(float* __restrict__ p, int n) {
    int i = blockIdx.x * blockDim.x + threadIdx.x;
    int stride = gridDim.x * blockDim.x;
    for (; i < n; i += stride) p[i] = 0.0f;
}

// ---------------------------------------------------------------------------
// Edge scatter: one wave32 per edge, fully unrolled over D/32 column chunks.
//   msg[dst][c] += feat[src][c]     (coalesced 128B atomic bursts)
//   deg[dst]    += 1.0              (lane 0, first layer only)
// Prefetches the next edge's source row (global_prefetch_b8) to hide the
// random-gather latency; all buffers are L2-resident on MI455X (192 MB L2).
// ---------------------------------------------------------------------------
template <int D, int ADD_DEG>
__global__ void edge_scatter(const float* __restrict__ feat,
                             const int*   __restrict__ src,
                             const int*   __restrict__ dst,
                             float* __restrict__ msg,
                             float* __restrict__ deg,
                             int nEdges) {
    int gtid  = blockIdx.x * blockDim.x + threadIdx.x;
    int warp  = gtid >> 5;
    int lane  = threadIdx.x & 31;
    int nWarp = (gridDim.x * blockDim.x) >> 5;
    constexpr int CHUNKS = D / 32;

    for (int e = warp; e < nEdges; e += nWarp) {
        int s = src[e];
        int d = dst[e];

        // Prefetch next iteration's source row; lanes spread across the row
        // so every cacheline of the D*4-byte row is touched.
        int eNext = e + nWarp;
        if (eNext < nEdges) {
            int sn = src[eNext];
            __builtin_prefetch(feat + (size_t)sn * D + lane * CHUNKS, 0, 0);
        }

        const float* fs = feat + (size_t)s * D;
        float*       md = msg  + (size_t)d * D;
#pragma unroll
        for (int c = 0; c < CHUNKS; ++c) {
            float v = fs[lane + 32 * c];
            atomic_add_f32_dev(md + lane + 32 * c, v);
        }
        if (ADD_DEG && lane == 0)
            atomic_add_f32_dev(deg + d, 1.0f);
    }
}

// ---------------------------------------------------------------------------
// Fused SAGEConv GEMM + bias + ReLU using V_WMMA_F32_16X16X4_F32 (wave32).
//   out = relu( (msg/deg) @ w_l^T + b_l + self @ w_r^T )
// One wave computes a 16(M) x 16(N) f32 tile; both GEMMs chain into one
// accumulator. A-frag: lane l -> row (l&15), K-pair base 2*(l>>4).
// B-frag: lane l -> col (l&15) of W^T == row of W, same K-pair indexing.
// C/D: VGPR i -> row i + 8*(l>>4), col (l&15).
// ---------------------------------------------------------------------------
template <int K>
__global__ void sage_gemm(const float* __restrict__ msg,   // [M,K] neighbor sums
                          const float* __restrict__ self,  // [M,K] self features
                          const float* __restrict__ deg,   // [M]
                          const float* __restrict__ w_l,   // [Ntot,K] row-major
                          const float* __restrict__ b_l,   // [Ntot]
                          const float* __restrict__ w_r,   // [Ntot,K] row-major
                          float* __restrict__ out,         // [M,Ntot]
                          int M, int Ntot) {
    int wid  = (blockIdx.x * blockDim.x + threadIdx.x) >> 5;  // wave-uniform
    int lane = threadIdx.x & 31;
    int nTilesN = Ntot >> 4;
    int mTile = wid / nTilesN;
    int nTile = wid - mTile * nTilesN;
    if (mTile * 16 >= M) return;  // wave-uniform: EXEC stays all-1s for WMMA

    int rsel = lane & 15;          // A row / B column / C column selector
    int kh   = (lane >> 4) << 1;   // K-pair base: 0 or 2
    int mA   = mTile * 16 + rsel;
    int nB   = nTile * 16 + rsel;

    float invd = 1.0f / fmaxf(deg[mA], 1.0f);
    const float* msgRow = msg  + (size_t)mA * K;
    const float* slfRow = self + (size_t)mA * K;
    const float* wlRow  = w_l  + (size_t)nB * K;
    const float* wrRow  = w_r  + (size_t)nB * K;

    v8f acc = {};
#pragma unroll 4
    for (int k0 = 0; k0 < K; k0 += 4) {
        int ka = k0 + kh;  // even -> 8B-aligned v2f loads
        v2f a1 = *(const v2f*)(msgRow + ka);
        a1 = a1 * invd;                              // fold mean into A-frag
        v2f b1 = *(const v2f*)(wlRow + ka);
        acc = __builtin_amdgcn_wmma_f32_16x16x4_f32(
            false, a1, false, b1, (short)0, acc, false, false);
        v2f a2 = *(const v2f*)(slfRow + ka);
        v2f b2 = *(const v2f*)(wrRow + ka);
        acc = __builtin_amdgcn_wmma_f32_16x16x4_f32(
            false, a2, false, b2, (short)0, acc, false, false);
    }

    float bias  = b_l[nB];
    int   mBase = mTile * 16 + ((lane >> 4) << 3);   // +8 for upper half-wave
#pragma unroll
    for (int i = 0; i < 8; ++i) {
        float v = acc[i] + bias;
        out[(size_t)(mBase + i) * Ntot + nB] = fmaxf(v, 0.0f);
    }
}

// ---------------------------------------------------------------------------
// Host-side launch. Inputs (setup_inputs order):
//  0:x[50000,128] f32  1:edge_index[2,800000] int  2:w_l1[256,128]  3:b_l1[256]
//  4:w_r1[256,128]  5:w_l2[256,256]  6:b_l2[256]  7:w_r2[256,256]
// Output: [50000,256] f32.
// ---------------------------------------------------------------------------
extern "C" void kernel_launch(void* const* d_in, const int* in_sizes, int n_in,
                              void* d_out, int out_size, void* d_ws, size_t ws_size,
                              hipStream_t stream) {
    (void)in_sizes; (void)n_in; (void)out_size; (void)ws_size;

    const float* x    = (const float*)d_in[0];
    const int*   ei   = (const int*)  d_in[1];
    const int*   src  = ei;
    const int*   dst  = ei + N_EDGES;
    const float* w_l1 = (const float*)d_in[2];
    const float* b_l1 = (const float*)d_in[3];
    const float* w_r1 = (const float*)d_in[4];
    const float* w_l2 = (const float*)d_in[5];
    const float* b_l2 = (const float*)d_in[6];
    const float* w_r2 = (const float*)d_in[7];
    float*       out  = (float*)d_out;

    // Workspace layout: [deg | msg(50000x256) | h(50000x256)] ~ 103 MB.
    char*  ws   = (char*)d_ws;
    float* deg  = (float*)ws;
    size_t off1 = ((size_t)N_NODES * sizeof(float) + 255) & ~(size_t)255;
    float* msg  = (float*)(ws + off1);
    size_t off2 = off1 + (size_t)N_NODES * D_HID * sizeof(float);
    float* h    = (float*)(ws + off2);

    const int TPB = 256;  // 8 wave32s per block

    // ---- Layer 1 ----
    zero_f32<<<1024, TPB, 0, stream>>>(deg, N_NODES);
    zero_f32<<<4096, TPB, 0, stream>>>(msg, N_NODES * D_IN);
    edge_scatter<D_IN, 1><<<4096, TPB, 0, stream>>>(x, src, dst, msg, deg,
                                                    N_EDGES);
    {
        int waves  = (N_NODES / 16) * (D_HID / 16);  // 50000
        int blocks = waves / (TPB / 32);             // 6250
        sage_gemm<D_IN><<<blocks, TPB, 0, stream>>>(msg, x, deg,
                                                    w_l1, b_l1, w_r1,
                                                    h, N_NODES, D_HID);
    }

    // ---- Layer 2 (reuse msg buffer, now 256-wide; deg unchanged) ----
    zero_f32<<<8192, TPB, 0, stream>>>(msg, N_NODES * D_HID);
    edge_scatter<D_HID, 0><<<4096, TPB, 0, stream>>>(h, src, dst, msg, deg,
                                                     N_EDGES);
    {
        int waves  = (N_NODES / 16) * (D_OUT / 16);
        int blocks = waves / (TPB / 32);
        sage_gemm<D_HID><<<blocks, TPB, 0, stream>>>(msg, h, deg,
                                                     w_l2, b_l2, w_r2,
                                                     out, N_NODES, D_OUT);
    }
}